// LaneATTTime_352187319115
// MI455X (gfx1250) — compile-verified
//
#include <hip/hip_runtime.h>
#include <hip/hip_bf16.h>
#include <stdint.h>

// ---- problem constants (from reference) ----
#define A_N    2784        // anchors: 72*6 + 128*15 + 72*6
#define D_N    704         // C_AF * FMAP_H = 64*11
#define B_N    8
#define T_N    4
#define H_N    11
#define W_N    20
#define HW_N   220
#define CIN    512
#define CAF    64
#define HEAD_K 3520        // 5 * 704  (past0..2 | att_feat | cur)
#define SC_N   2783        // A_N - 1 (valid attention cols)
#define SC_P   2816        // padded stride (= 44*64) for 64-wide N blocking
#define NH_P   80          // padded head N (cls2 + reg74 -> 80)
#define OUT_C  77

typedef __attribute__((ext_vector_type(2))) float v2f;
typedef __attribute__((ext_vector_type(8))) float v8f;

// One fp32 WMMA step: D(16x16) += A(16x4) * B(4x16)
__device__ __forceinline__ v8f wmma4(v8f acc, float a0, float a1, float b0, float b1) {
    v2f a; a[0] = a0; a[1] = a1;
    v2f b; b[0] = b0; b[1] = b1;
    return __builtin_amdgcn_wmma_f32_16x16x4_f32(
        false, a, false, b, (short)0, acc, false, false);
}

#define V8F_ZERO {0.f,0.f,0.f,0.f,0.f,0.f,0.f,0.f}

// ------------------------------------------------------------------
// Pack att_w (704 x 2783) -> attw_p (704 x 2816, zero-padded)
// and att_b -> attb_p (2816)
// ------------------------------------------------------------------
__global__ void k_pack_attw(const float* __restrict__ att_w, const float* __restrict__ att_b,
                            float* __restrict__ attw_p, float* __restrict__ attb_p) {
    const size_t idx = (size_t)blockIdx.x * blockDim.x + threadIdx.x;
    const size_t total = (size_t)D_N * SC_P;
    if (idx >= total) return;
    const int n = (int)(idx % SC_P);
    const int k = (int)(idx / SC_P);
    attw_p[idx] = (n < SC_N) ? att_w[(size_t)k * SC_N + n] : 0.f;
    if (idx < SC_P) attb_p[idx] = (idx < SC_N) ? att_b[idx] : 0.f;
}

// ------------------------------------------------------------------
// Pack [cls_w | reg_w] -> wcat (3520 x 80), biases -> bcat (80)
// ------------------------------------------------------------------
__global__ void k_pack_head(const float* __restrict__ cls_w, const float* __restrict__ cls_b,
                            const float* __restrict__ reg_w, const float* __restrict__ reg_b,
                            float* __restrict__ wcat, float* __restrict__ bcat) {
    const size_t idx = (size_t)blockIdx.x * blockDim.x + threadIdx.x;
    const size_t total = (size_t)HEAD_K * NH_P;
    if (idx >= total) return;
    const int n = (int)(idx % NH_P);
    const int k = (int)(idx / NH_P);
    float v = 0.f;
    if (n < 2)       v = cls_w[(size_t)k * 2 + n];
    else if (n < 76) v = reg_w[(size_t)k * 74 + (n - 2)];
    wcat[idx] = v;
    if (idx < NH_P) {
        float bv = 0.f;
        if (idx < 2) bv = cls_b[idx];
        else if (idx < 76) bv = reg_b[idx - 2];
        bcat[idx] = bv;
    }
}

// ------------------------------------------------------------------
// 1x1 conv: feats[bt,o,hw] = sum_c w1[o,c]*x[bt,c,hw] + b1[o]
// wave computes 32x32 tile (2x2); loads clamped, stores guarded
// ------------------------------------------------------------------
__global__ void k_conv(const float* __restrict__ x, const float* __restrict__ w1,
                       const float* __restrict__ b1, float* __restrict__ feats) {
    const int lane = threadIdx.x & 31;
    int wave = blockIdx.x * (blockDim.x >> 5) + (threadIdx.x >> 5);
    const int NT2 = 7;                       // ceil(220/32)
    const int nt = wave % NT2; wave /= NT2;
    const int mt = wave % 2;  wave /= 2;     // 64/32
    const int bt = wave;
    if (bt >= B_N * T_N) return;
    const int l15 = lane & 15;
    const int kh  = (lane >> 4) << 1;
    const int m0 = mt * 32 + l15;
    const int n0 = nt * 32 + l15;
    const int n1 = n0 + 16;
    const int n0c = (n0 < HW_N) ? n0 : (HW_N - 1);
    const int n1c = (n1 < HW_N) ? n1 : (HW_N - 1);
    const float* xb  = x + (size_t)bt * CIN * HW_N;
    const float* wa0 = w1 + (size_t)m0 * CIN;
    const float* wa1 = wa0 + (size_t)16 * CIN;
    v8f acc00 = V8F_ZERO, acc01 = V8F_ZERO, acc10 = V8F_ZERO, acc11 = V8F_ZERO;
    for (int k = 0; k < CIN; k += 4) {
        const int ka = k + kh;
        const float a00 = wa0[ka], a01 = wa0[ka + 1];
        const float a10 = wa1[ka], a11 = wa1[ka + 1];
        const float* xr0 = xb + (size_t)ka * HW_N;
        const float* xr1 = xr0 + HW_N;
        const float b00 = xr0[n0c], b01 = xr1[n0c];
        const float b10 = xr0[n1c], b11 = xr1[n1c];
        acc00 = wmma4(acc00, a00, a01, b00, b01);
        acc01 = wmma4(acc01, a00, a01, b10, b11);
        acc10 = wmma4(acc10, a10, a11, b00, b01);
        acc11 = wmma4(acc11, a10, a11, b10, b11);
    }
    const int mb0 = mt * 32 + ((lane >> 4) ? 8 : 0);
    const int mb1 = mb0 + 16;
    #pragma unroll
    for (int r = 0; r < 8; ++r) {
        const float bia0 = b1[mb0 + r], bia1 = b1[mb1 + r];
        float* f0 = feats + ((size_t)bt * CAF + mb0 + r) * HW_N;
        float* f1 = feats + ((size_t)bt * CAF + mb1 + r) * HW_N;
        if (n0 < HW_N) { f0[n0] = acc00[r] + bia0; f1[n0] = acc10[r] + bia1; }
        if (n1 < HW_N) { f0[n1] = acc01[r] + bia0; f1[n1] = acc11[r] + bia1; }
    }
}

// ------------------------------------------------------------------
// Gather rois -> head_in rows of 3520 (branch-free: load then select)
// also writes out[:, :, 2] = anchors[:, 2]
// ------------------------------------------------------------------
__global__ void k_gather(const float* __restrict__ feats, const int* __restrict__ cut_xs,
                         const unsigned char* __restrict__ invalid,
                         const float* __restrict__ anchors,
                         float* __restrict__ head_in, float* __restrict__ out) {
    size_t idx = (size_t)blockIdx.x * blockDim.x + threadIdx.x;
    const size_t total = (size_t)B_N * A_N * T_N * D_N;
    if (idx >= total) return;
    const int d = (int)(idx % D_N); size_t r = idx / D_N;
    const int t = (int)(r % T_N); r /= T_N;
    const int a = (int)(r % A_N);
    const int b = (int)(r / A_N);
    const int c = d / H_N, h = d % H_N;
    const int xcol = cut_xs[a * H_N + h];              // pre-clamped to [0,19]
    float v = feats[(((size_t)(b * T_N + t)) * CAF + c) * HW_N + h * W_N + xcol];
    v = invalid[a * H_N + h] ? 0.f : v;                // value select, no branch
    const int slot = (t == T_N - 1) ? 4 : t;           // cur -> slot 4
    head_in[((size_t)b * A_N + a) * HEAD_K + (size_t)slot * D_N + d] = v;
    if (t == 0 && d == 0)
        out[((size_t)b * A_N + a) * OUT_C + 2] = anchors[(size_t)a * OUT_C + 2];
}

// ------------------------------------------------------------------
// scores = cur(22272 x 704) @ attw_p(704 x 2816) + attb_p
// wave computes 48x64 (3x4 WMMA tiles, 12 accumulators); guard-free
// ------------------------------------------------------------------
__global__ void k_scores(const float* __restrict__ head_in, const float* __restrict__ attw_p,
                         const float* __restrict__ attb_p, float* __restrict__ scores) {
    const int lane = threadIdx.x & 31;
    const int wave = blockIdx.x * (blockDim.x >> 5) + (threadIdx.x >> 5);
    const int MT = (B_N * A_N) / 48;         // 464
    const int NT = SC_P / 64;                // 44
    if (wave >= MT * NT) return;
    const int nt = wave % NT;
    const int mt = wave / NT;
    const int l15 = lane & 15;
    const int kh  = (lane >> 4) << 1;
    const int m0 = mt * 48 + l15;
    const int n0 = nt * 64 + l15;
    const float* cu0 = head_in + (size_t)m0 * HEAD_K + 4 * D_N;
    const float* cu1 = cu0 + (size_t)16 * HEAD_K;
    const float* cu2 = cu0 + (size_t)32 * HEAD_K;
    v8f acc[12];
    #pragma unroll
    for (int q = 0; q < 12; ++q) acc[q] = (v8f)V8F_ZERO;
    for (int k = 0; k < D_N; k += 4) {
        const int ka = k + kh;
        float a0[3], a1[3];
        a0[0] = cu0[ka]; a1[0] = cu0[ka + 1];
        a0[1] = cu1[ka]; a1[1] = cu1[ka + 1];
        a0[2] = cu2[ka]; a1[2] = cu2[ka + 1];
        const float* w0 = attw_p + (size_t)ka * SC_P + n0;
        const float* w1 = w0 + SC_P;
        float b0[4], b1[4];
        #pragma unroll
        for (int j = 0; j < 4; ++j) { b0[j] = w0[j * 16]; b1[j] = w1[j * 16]; }
        #pragma unroll
        for (int i = 0; i < 3; ++i)
            #pragma unroll
            for (int j = 0; j < 4; ++j)
                acc[i * 4 + j] = wmma4(acc[i * 4 + j], a0[i], a1[i], b0[j], b1[j]);
    }
    const int mb = mt * 48 + ((lane >> 4) ? 8 : 0);
    #pragma unroll
    for (int i = 0; i < 3; ++i) {
        #pragma unroll
        for (int j = 0; j < 4; ++j) {
            const int n = n0 + j * 16;
            const float bias = attb_p[n];
            #pragma unroll
            for (int r = 0; r < 8; ++r)
                scores[(size_t)(mb + i * 16 + r) * SC_P + n] = acc[i * 4 + j][r] + bias;
        }
    }
}

// ------------------------------------------------------------------
// row softmax in place (2783 valid cols, stride 2816)
// ------------------------------------------------------------------
__global__ void k_softmax(float* __restrict__ scores) {
    __shared__ float red[256];
    float* p = scores + (size_t)blockIdx.x * SC_P;
    float mx = -3.4e38f;
    for (int i = threadIdx.x; i < SC_N; i += blockDim.x) mx = fmaxf(mx, p[i]);
    red[threadIdx.x] = mx; __syncthreads();
    for (int s = blockDim.x >> 1; s > 0; s >>= 1) {
        if ((int)threadIdx.x < s) red[threadIdx.x] = fmaxf(red[threadIdx.x], red[threadIdx.x + s]);
        __syncthreads();
    }
    mx = red[0]; __syncthreads();
    float sum = 0.f;
    for (int i = threadIdx.x; i < SC_N; i += blockDim.x) {
        const float e = __expf(p[i] - mx); p[i] = e; sum += e;
    }
    red[threadIdx.x] = sum; __syncthreads();
    for (int s = blockDim.x >> 1; s > 0; s >>= 1) {
        if ((int)threadIdx.x < s) red[threadIdx.x] += red[threadIdx.x + s];
        __syncthreads();
    }
    const float inv = 1.f / red[0];
    for (int i = threadIdx.x; i < SC_N; i += blockDim.x) p[i] *= inv;
}

// ------------------------------------------------------------------
// att_feat[b] = M[b] @ cur[b];  M[i][j] = (j==i) ? 0 : att[b,i,j-(j>i)]
// diagonal handled by value-select; padded stride keeps gathers in-bounds.
// wave computes 48x64 (3x4 tiles); result -> head_in slot 3; guard-free
// ------------------------------------------------------------------
__global__ void k_attfeat(const float* __restrict__ att, float* head_in) {
    const int lane = threadIdx.x & 31;
    int wave = blockIdx.x * (blockDim.x >> 5) + (threadIdx.x >> 5);
    const int IT = A_N / 48;                 // 58
    const int NT = D_N / 64;                 // 11
    if (wave >= B_N * IT * NT) return;
    const int nt = wave % NT; wave /= NT;
    const int it = wave % IT;
    const int b  = wave / IT;
    const int l15 = lane & 15;
    const int kh  = (lane >> 4) << 1;
    const int n0 = nt * 64 + l15;
    int irow[3];
    const float* arow[3];
    #pragma unroll
    for (int i = 0; i < 3; ++i) {
        irow[i] = it * 48 + i * 16 + l15;
        arow[i] = att + ((size_t)b * A_N + irow[i]) * SC_P;
    }
    const float* curb = head_in + (size_t)b * A_N * HEAD_K + 4 * D_N;
    v8f acc[12];
    #pragma unroll
    for (int q = 0; q < 12; ++q) acc[q] = (v8f)V8F_ZERO;
    for (int k = 0; k < A_N; k += 4) {
        const int j0 = k + kh, j1 = j0 + 1;
        float a0[3], a1[3];
        #pragma unroll
        for (int i = 0; i < 3; ++i) {
            float t0 = arow[i][j0 - (j0 > irow[i])];
            float t1 = arow[i][j1 - (j1 > irow[i])];
            a0[i] = (j0 == irow[i]) ? 0.f : t0;
            a1[i] = (j1 == irow[i]) ? 0.f : t1;
        }
        const float* c0 = curb + (size_t)j0 * HEAD_K + n0;
        const float* c1 = curb + (size_t)j1 * HEAD_K + n0;
        float b0[4], b1[4];
        #pragma unroll
        for (int j = 0; j < 4; ++j) { b0[j] = c0[j * 16]; b1[j] = c1[j * 16]; }
        #pragma unroll
        for (int i = 0; i < 3; ++i)
            #pragma unroll
            for (int j = 0; j < 4; ++j)
                acc[i * 4 + j] = wmma4(acc[i * 4 + j], a0[i], a1[i], b0[j], b1[j]);
    }
    const int ib = it * 48 + ((lane >> 4) ? 8 : 0);
    #pragma unroll
    for (int i = 0; i < 3; ++i) {
        #pragma unroll
        for (int r = 0; r < 8; ++r) {
            float* o = head_in + ((size_t)b * A_N + ib + i * 16 + r) * HEAD_K + 3 * D_N + n0;
            #pragma unroll
            for (int j = 0; j < 4; ++j)
                o[j * 16] = acc[i * 4 + j][r];
        }
    }
}

// ------------------------------------------------------------------
// head GEMM: out76 = head_in(22272 x 3520) @ wcat(3520 x 80) + bcat
// wave computes 32x80 (2x5 tiles); epilogue scatters into 77-wide
// output with anchor add
// ------------------------------------------------------------------
__global__ void k_head(const float* __restrict__ head_in, const float* __restrict__ wcat,
                       const float* __restrict__ bcat, const float* __restrict__ anchors,
                       float* __restrict__ out) {
    const int lane = threadIdx.x & 31;
    const int wave = blockIdx.x * (blockDim.x >> 5) + (threadIdx.x >> 5);
    const int MT2 = (B_N * A_N) / 32;        // 696
    if (wave >= MT2) return;
    const int l15 = lane & 15;
    const int kh  = (lane >> 4) << 1;
    const int m0 = wave * 32 + l15;
    const float* hi0 = head_in + (size_t)m0 * HEAD_K;
    const float* hi1 = hi0 + (size_t)16 * HEAD_K;
    v8f acc[10];
    #pragma unroll
    for (int j = 0; j < 10; ++j) acc[j] = (v8f)V8F_ZERO;
    for (int k = 0; k < HEAD_K; k += 4) {
        const int ka = k + kh;
        const float a00 = hi0[ka], a01 = hi0[ka + 1];
        const float a10 = hi1[ka], a11 = hi1[ka + 1];
        const float* w0 = wcat + (size_t)ka * NH_P + l15;
        const float* w1 = w0 + NH_P;
        #pragma unroll
        for (int j = 0; j < 5; ++j) {
            const float b0 = w0[j * 16], b1 = w1[j * 16];
            acc[j]     = wmma4(acc[j],     a00, a01, b0, b1);
            acc[5 + j] = wmma4(acc[5 + j], a10, a11, b0, b1);
        }
    }
    const int mb0 = wave * 32 + ((lane >> 4) ? 8 : 0);
    #pragma unroll
    for (int j = 0; j < 5; ++j) {
        const int n = j * 16 + l15;
        if (n >= 76) continue;               // lane guard (only hits in j==4)
        const float bias = bcat[n];
        const int col = (n < 2) ? n : (n + 1);
        #pragma unroll
        for (int r = 0; r < 8; ++r) {
            const int mm0 = mb0 + r, mm1 = mb0 + 16 + r;
            const int a0i = mm0 % A_N, a1i = mm1 % A_N;
            float v0 = acc[j][r] + bias;
            float v1 = acc[5 + j][r] + bias;
            if (n >= 2) {
                v0 += anchors[(size_t)a0i * OUT_C + col];
                v1 += anchors[(size_t)a1i * OUT_C + col];
            }
            out[(size_t)mm0 * OUT_C + col] = v0;
            out[(size_t)mm1 * OUT_C + col] = v1;
        }
    }
}

// ------------------------------------------------------------------
extern "C" void kernel_launch(void* const* d_in, const int* in_sizes, int n_in,
                              void* d_out, int out_size, void* d_ws, size_t ws_size,
                              hipStream_t stream) {
    const float* x       = (const float*)d_in[0];
    const float* conv1_w = (const float*)d_in[1];
    const float* conv1_b = (const float*)d_in[2];
    const float* att_w   = (const float*)d_in[3];
    const float* att_b   = (const float*)d_in[4];
    const float* cls_w   = (const float*)d_in[5];
    const float* cls_b   = (const float*)d_in[6];
    const float* reg_w   = (const float*)d_in[7];
    const float* reg_b   = (const float*)d_in[8];
    const float* anchors = (const float*)d_in[9];
    const int*   cut_xs  = (const int*)d_in[10];
    const unsigned char* invalid = (const unsigned char*)d_in[11];  // jnp bool -> byte
    float* out = (float*)d_out;

    // workspace layout (floats)
    float* feats   = (float*)d_ws;                                   //    450,560
    float* head_in = feats   + (size_t)B_N * T_N * CAF * HW_N;       // 78,397,440
    float* scores  = head_in + (size_t)B_N * A_N * HEAD_K;           // 62,717,952 (stride 2816)
    float* attw_p  = scores  + (size_t)B_N * A_N * SC_P;             //  1,982,464
    float* attb_p  = attw_p  + (size_t)D_N * SC_P;                   //      2,816
    float* wcat    = attb_p  + SC_P;                                 //    281,600
    float* bcat    = wcat    + (size_t)HEAD_K * NH_P;                //         80

    // weight packing (independent of data path)
    k_pack_attw<<<(D_N * SC_P + 255) / 256, 256, 0, stream>>>(att_w, att_b, attw_p, attb_p);
    k_pack_head<<<(HEAD_K * NH_P + 255) / 256, 256, 0, stream>>>(cls_w, cls_b, reg_w, reg_b, wcat, bcat);

    // 1) conv: 32 * 2 * 7 = 448 waves -> 56 blocks of 8 waves
    k_conv<<<56, 256, 0, stream>>>(x, conv1_w, conv1_b, feats);

    // 2) gather: B*A*4*704 threads
    {
        const size_t total = (size_t)B_N * A_N * T_N * D_N;
        k_gather<<<(int)((total + 255) / 256), 256, 0, stream>>>(feats, cut_xs, invalid, anchors, head_in, out);
    }

    // 3) scores GEMM: 464 * 44 = 20,416 waves -> 2,552 blocks
    k_scores<<<2552, 256, 0, stream>>>(head_in, attw_p, attb_p, scores);

    // 4) softmax: one block per row
    k_softmax<<<B_N * A_N, 256, 0, stream>>>(scores);

    // 5) att_feat GEMM: 8 * 58 * 11 = 5,104 waves -> 638 blocks
    k_attfeat<<<638, 256, 0, stream>>>(scores, head_in);

    // 6) head GEMM: 696 waves -> 87 blocks
    k_head<<<87, 256, 0, stream>>>(head_in, wcat, bcat, anchors, out);
}